// Involution_20624432955940
// MI455X (gfx1250) — compile-verified
//
#include <hip/hip_runtime.h>
#include <hip/hip_bf16.h>

// ---------------------------------------------------------------------------
// Involution (K=3,S=1) + BN + LeakyReLU on MI455X (gfx1250).
//
// Math collapse: yout[b,p,q,c] = in[b,p,q,c] * Sker[b,p,q],
//   Sker[b,p,q] = sum_{ki,kj in [0,3), 0<=p-ki<126, 0<=q-kj<126}
//                   kernel[b, p-ki, q-kj, ki*3+kj]
//   kernel[b,i,j,t] = in[b,i+1,j+1,:] . W1[:,t] + b1[t]
//
// Memory-bound (~160MB traffic @ 23.3TB/s => ~7us floor). The kernel-field
// GEMM ((16*126*126) x 9 with K=64) uses V_WMMA_F32_16X16X4_F32: one wave
// computes a 16-pixel x 16-tap (9 used, zero-padded) tile in 16 k-steps of 4
// channels. B operands are staged in LDS (padded+transposed W1) so the inner
// loop is branch-free: global_load_b64 (A) + ds_load_b64 (B) + v_wmma.
// ---------------------------------------------------------------------------

typedef __attribute__((ext_vector_type(2))) float v2f;
typedef __attribute__((ext_vector_type(8))) float v8f;

#define B_   16
#define H_   128
#define W_   128
#define C_   64
#define NH   126
#define NW   126
#define NPIX (B_ * H_ * W_)            // 262144 output pixels
#define NCTR (B_ * NH * NW)            // 254016 center pixels
#define NTILE (NCTR / 16)              // 15876 WMMA tiles (exact)
#define NBLK_RED 1024                  // partial-reduction blocks

// ---------------------------------------------------------------------------
// Pass 1: kernel field via WMMA.  One wave per 16-pixel tile.
// K9 layout: [center_pixel][16] (taps 9..15 are zero padding).
// ---------------------------------------------------------------------------
__global__ void __launch_bounds__(256)
invol_field_wmma(const float* __restrict__ in,
                 const float* __restrict__ W1,   // (64, 9) row-major
                 const float* __restrict__ b1,   // (9,)
                 float* __restrict__ K9) {       // (NCTR, 16)
  // Stage padded+transposed weights in LDS: W1t[n][k], n=tap (16, zero-padded
  // past 9), k=channel (64). Lane B-operand reads become one ds_load_b64.
  __shared__ float W1t[16 * 64];
  __shared__ float b1p[16];
  for (int idx = threadIdx.x; idx < 16 * 64; idx += 256) {
    const int n = idx >> 6;
    const int k = idx & 63;
    W1t[idx] = (n < 9) ? W1[k * 9 + n] : 0.0f;
  }
  if (threadIdx.x < 16)
    b1p[threadIdx.x] = (threadIdx.x < 9) ? b1[threadIdx.x] : 0.0f;
  __syncthreads();

  const int wave = blockIdx.x * (blockDim.x >> 5) + (threadIdx.x >> 5);
  if (wave < NTILE) {                            // wave-uniform guard
    const int lane = threadIdx.x & 31;
    const int m = lane & 15;                     // A-row / pixel-in-tile
    const int h = lane >> 4;                     // half-wave (K split)
    const int n = m;                             // B/D column (tap)

    // Decode this lane's A pixel -> address of its channel vector (center).
    const int P = wave * 16 + m;
    const int b = P / (NH * NW);
    const int rem = P - b * (NH * NW);
    const int i = rem / NW;
    const int j = rem - i * NW;
    const float* cptr =
        in + (((b * H_ + (i + 1)) * W_) + (j + 1)) * C_ + 2 * h;
    const float* wrow = &W1t[n * 64 + 2 * h];

    const float b1n = b1p[n];
    v8f acc;
#pragma unroll
    for (int r = 0; r < 8; ++r) acc[r] = b1n;    // fold bias into C

#pragma unroll
    for (int c0 = 0; c0 < C_; c0 += 4) {
      // A: 16x4 fp32. lane(m,h): VGPR0 = A[m, c0+2h], VGPR1 = A[m, c0+2h+1]
      v2f a = *(const v2f*)(cptr + c0);
      // B: 4x16 fp32 from LDS (already padded) -- branch-free ds_load_b64
      v2f bb = *(const v2f*)(wrow + c0);
      acc = __builtin_amdgcn_wmma_f32_16x16x4_f32(
          false, a, false, bb, (short)0, acc, false, false);
    }

    // D layout: VGPR r, lane(n,h) -> pixel (wave*16 + r + 8h), tap n.
    const int base = wave * 16 + 8 * h;
#pragma unroll
    for (int r = 0; r < 8; ++r) {
      K9[(base + r) * 16 + n] = acc[r];          // 2x 64B coalesced segments
    }
  }
}

// ---------------------------------------------------------------------------
// Pass 2: Sker[b,p,q] = sum of up-to-9 neighboring kernel-field taps.
// ---------------------------------------------------------------------------
__global__ void __launch_bounds__(256)
invol_sker(const float* __restrict__ K9, float* __restrict__ Sker) {
  const int idx = blockIdx.x * blockDim.x + threadIdx.x;
  if (idx >= NPIX) return;
  const int b = idx >> 14;          // / (128*128)
  const int rem = idx & 16383;
  const int p = rem >> 7;
  const int q = rem & 127;
  float s = 0.0f;
#pragma unroll
  for (int ki = 0; ki < 3; ++ki) {
    const int i = p - ki;
    if ((unsigned)i < (unsigned)NH) {
#pragma unroll
      for (int kj = 0; kj < 3; ++kj) {
        const int j = q - kj;
        if ((unsigned)j < (unsigned)NW) {
          s += K9[(((b * NH + i) * NW) + j) * 16 + (ki * 3 + kj)];
        }
      }
    }
  }
  Sker[idx] = s;
}

// ---------------------------------------------------------------------------
// Pass 3: per-block partial BN sums (deterministic two-level reduction).
// Block = 256 threads = 4 pixel-groups x 64 channels.
// ---------------------------------------------------------------------------
__global__ void __launch_bounds__(256)
invol_partials(const float* __restrict__ in, const float* __restrict__ Sker,
               float* __restrict__ psum, float* __restrict__ psumsq) {
  __shared__ float ls[256];
  __shared__ float lq[256];
  const int c = threadIdx.x & 63;
  const int g = threadIdx.x >> 6;                // 0..3
  const int PPB = NPIX / NBLK_RED;               // 256 pixels per block
  const int p0 = blockIdx.x * PPB;
  float s = 0.0f, q = 0.0f;
  for (int t = g; t < PPB; t += 4) {
    const int pix = p0 + t;
    const float y = in[pix * C_ + c] * Sker[pix];
    s += y;
    q += y * y;
  }
  ls[threadIdx.x] = s;
  lq[threadIdx.x] = q;
  __syncthreads();
  if (g == 0) {
    const float S = ls[c] + ls[c + 64] + ls[c + 128] + ls[c + 192];
    const float Q = lq[c] + lq[c + 64] + lq[c + 128] + lq[c + 192];
    psum[blockIdx.x * 64 + c] = S;
    psumsq[blockIdx.x * 64 + c] = Q;
  }
}

// ---------------------------------------------------------------------------
// Pass 4: fold partials -> per-channel affine (scale, shift).
// ---------------------------------------------------------------------------
__global__ void invol_finalize(const float* __restrict__ psum,
                               const float* __restrict__ psumsq,
                               const float* __restrict__ gamma,
                               const float* __restrict__ beta,
                               float* __restrict__ scale,
                               float* __restrict__ shift) {
  const int c = threadIdx.x;                     // 64 threads
  float s = 0.0f, q = 0.0f;
  for (int blk = 0; blk < NBLK_RED; ++blk) {
    s += psum[blk * 64 + c];
    q += psumsq[blk * 64 + c];
  }
  const float invN = 1.0f / (float)NPIX;
  const float mean = s * invN;
  const float var = q * invN - mean * mean;
  const float sc = gamma[c] * rsqrtf(var + 0.001f);
  scale[c] = sc;
  shift[c] = beta[c] - mean * sc;
}

// ---------------------------------------------------------------------------
// Pass 5: out = leaky(in * Sker * scale[c] + shift[c]), float4-vectorized.
// ---------------------------------------------------------------------------
__global__ void __launch_bounds__(256)
invol_out(const float* __restrict__ in, const float* __restrict__ Sker,
          const float* __restrict__ scale, const float* __restrict__ shift,
          float* __restrict__ out) {
  const int idx = blockIdx.x * blockDim.x + threadIdx.x;  // float4 units
  const int NQ = NPIX * (C_ / 4);
  if (idx >= NQ) return;
  const int pix = idx >> 4;
  const int c0 = (idx & 15) * 4;
  const float sk = Sker[pix];
  float4 v = ((const float4*)in)[idx];
  float4 r;
  r.x = v.x * sk * scale[c0 + 0] + shift[c0 + 0];
  r.y = v.y * sk * scale[c0 + 1] + shift[c0 + 1];
  r.z = v.z * sk * scale[c0 + 2] + shift[c0 + 2];
  r.w = v.w * sk * scale[c0 + 3] + shift[c0 + 3];
  r.x = (r.x >= 0.0f) ? r.x : 0.3f * r.x;
  r.y = (r.y >= 0.0f) ? r.y : 0.3f * r.y;
  r.z = (r.z >= 0.0f) ? r.z : 0.3f * r.z;
  r.w = (r.w >= 0.0f) ? r.w : 0.3f * r.w;
  ((float4*)out)[idx] = r;
}

// ---------------------------------------------------------------------------
extern "C" void kernel_launch(void* const* d_in, const int* in_sizes, int n_in,
                              void* d_out, int out_size, void* d_ws,
                              size_t ws_size, hipStream_t stream) {
  const float* in    = (const float*)d_in[0];   // (16,128,128,64)
  const float* W1    = (const float*)d_in[1];   // (64,9)
  const float* b1    = (const float*)d_in[2];   // (9,)
  const float* gamma = (const float*)d_in[3];   // (64,)
  const float* beta  = (const float*)d_in[4];   // (64,)
  float* out = (float*)d_out;

  // Workspace layout (floats)
  float* ws = (float*)d_ws;
  float* K9     = ws;                                  // NCTR*16 = 4,064,256
  float* Sker   = K9 + (size_t)NCTR * 16;              // NPIX    =   262,144
  float* psum   = Sker + NPIX;                         // NBLK_RED*64 = 65,536
  float* psumsq = psum + NBLK_RED * 64;                //             = 65,536
  float* scale  = psumsq + NBLK_RED * 64;              // 64
  float* shift  = scale + 64;                          // 64

  // Pass 1: WMMA kernel-field GEMM. 15876 waves, 8 waves/block.
  {
    const int waves_per_block = 256 / 32;
    const int blocks = (NTILE + waves_per_block - 1) / waves_per_block;
    invol_field_wmma<<<blocks, 256, 0, stream>>>(in, W1, b1, K9);
  }
  // Pass 2: Sker field.
  invol_sker<<<(NPIX + 255) / 256, 256, 0, stream>>>(K9, Sker);
  // Pass 3: BN partial sums (deterministic).
  invol_partials<<<NBLK_RED, 256, 0, stream>>>(in, Sker, psum, psumsq);
  // Pass 4: per-channel affine.
  invol_finalize<<<1, 64, 0, stream>>>(psum, psumsq, gamma, beta, scale, shift);
  // Pass 5: elementwise output.
  invol_out<<<(NPIX * (C_ / 4) + 255) / 256, 256, 0, stream>>>(in, Sker, scale,
                                                              shift, out);
}